// LundWeight_83090437308623
// MI455X (gfx1250) — compile-verified
//
#include <hip/hip_runtime.h>
#include <cstdint>

#define K_OSF      15.0f
#define K_EXPMAX   50.0f
#define K_AZERO    0.02f
#define K_AC       0.01f

__device__ __forceinline__ float fast_rcp(float x) {
    // v_rcp_f32 (~1 ulp) — plenty for weight tolerance here
    return __builtin_amdgcn_rcpf(x);
}

// zmax_calc with c == 1.0 (as in the reference call site). `a` is uniform
// per parameter set so the branches are scalar (no divergence).
__device__ __forceinline__ float zmax_calc(float a, float bb) {
    if (a < K_AZERO) {                       // a ~ 0
        return (1.0f > bb) ? bb : 1.0f;
    }
    if (fabsf(a - 1.0f) < K_AC) {            // a ~ c
        return bb / (bb + 1.0f);
    }
    float d  = bb - 1.0f;
    float zg = 0.5f * (bb + 1.0f - sqrtf(d * d + 4.0f * a * bb)) / (1.0f - a);
    if (zg > 0.9999f && bb > 100.0f) {
        zg = fminf(zg, 1.0f - a / bb);
    }
    return zg;
}

__global__ void __launch_bounds__(256)
lund_weight_kernel(const float* __restrict__ z,
                   const float* __restrict__ mT,
                   const int*   __restrict__ obs,     // observable (B,2), field 0 = mult
                   const float* __restrict__ pa,
                   const float* __restrict__ pb,
                   const float* __restrict__ pbase,
                   float* __restrict__ out,
                   int M, int S) {
    extern __shared__ float zsh[];           // M*S floats: staged z tile for this event
    __shared__ float wred[32];               // per-wave product partials

    const int b     = blockIdx.x;
    const int tid   = threadIdx.x;
    const int nt    = blockDim.x;
    const int total = M * S;

    const float* zev = z + (size_t)b * (size_t)total;

    // ---- Stage the event's z tile into LDS with CDNA5 async copies. ----
    // the LDS destination address comes from the VDST VGPR; low 32 bits of the generic pointer to
    // shared memory are exactly the wave-relative LDS byte address.
    uint32_t lds_base = (uint32_t)(uintptr_t)&zsh[0];
    const int nvec = total >> 2;             // 16-byte chunks
    for (int v = tid; v < nvec; v += nt) {
        uint32_t lo = lds_base + ((uint32_t)v << 4);
        uint32_t go = (uint32_t)v << 4;
        asm volatile("global_load_async_to_lds_b128 %0, %1, %2"
                     :: "v"(lo), "v"(go), "s"(zev) : "memory");
    }
    for (int e = (nvec << 2) + tid; e < total; e += nt) {   // tail (none for S=25,M=128)
        uint32_t lo = lds_base + ((uint32_t)e << 2);
        uint32_t go = (uint32_t)e << 2;
        asm volatile("global_load_async_to_lds_b32 %0, %1, %2"
                     :: "v"(lo), "v"(go), "s"(zev) : "memory");
    }

    // ---- Overlap: per-row (m) tables in registers while async copies fly. ----
    const float ap = pa[0];
    const float bp = pb[0];
    const float a0 = pbase[0];
    const float b0 = pbase[1];
    const float aeffp = (ap < K_AZERO) ? 0.0f : ap;   // reference drops a-term when a<0.02
    const float aeff0 = (a0 < K_AZERO) ? 0.0f : a0;
    const int   mult  = obs[2 * b];

    const int  m        = tid;
    const bool rowvalid = (m < M) && (m < mult);

    float bxp = 0.0f, bx0 = 0.0f, c1p = 0.0f, c10 = 0.0f;
    if (m < M) {
        float mtv = mT[(size_t)b * (size_t)M + m];
        float mt2 = mtv * mtv;
        bxp = bp * mt2;
        bx0 = b0 * mt2;
        float zmp = zmax_calc(ap, bxp);
        float zm0 = zmax_calc(a0, bx0);
        // c1 = b_exp/z_max + log(z_max) - a*log(1 - z_max)   (guard a==0 vs z_max==1)
        c1p = bxp * fast_rcp(zmp) + __logf(zmp)
              - ((aeffp != 0.0f) ? aeffp * __logf(1.0f - zmp) : 0.0f);
        c10 = bx0 * fast_rcp(zm0) + __logf(zm0)
              - ((aeff0 != 0.0f) ? aeff0 * __logf(1.0f - zm0) : 0.0f);
    }

    // ---- Wait for this wave's async copies, then sync the block (each thread
    //      consumes a row staged by *other* threads). ----
    asm volatile("s_wait_asynccnt 0" ::: "memory");
    __syncthreads();

    // ---- Walk own row: stride-25-word lane pattern is LDS bank-conflict-free. ----
    float prod = 1.0f;
    if (rowvalid) {
        const float* zrow = &zsh[m * S];
        for (int r = 0; r < S; ++r) {
            float zv = zrow[r];
            if (zv != 0.0f) {
                float rz  = fast_rcp(zv);
                float lz  = __logf(zv);
                float l1z = __logf(1.0f - zv);
                float fp = c1p - bxp * rz - lz + aeffp * l1z;
                float f0 = c10 - bx0 * rz - lz + aeff0 * l1z;
                fp = fminf(fmaxf(fp, -K_EXPMAX), K_EXPMAX);
                f0 = fminf(fmaxf(f0, -K_EXPMAX), K_EXPMAX);
                float Lp = __expf(fp);
                float L0 = __expf(f0);
                float w  = (r == 0) ? (Lp * fast_rcp(L0))
                                    : ((K_OSF - Lp) * fast_rcp(K_OSF - L0));
                prod *= w;
            }
        }
    }

    // ---- wave32 product reduction, then cross-wave via LDS. ----
    #pragma unroll
    for (int off = 16; off > 0; off >>= 1) {
        prod *= __shfl_xor(prod, off, 32);
    }
    const int lane = tid & 31;
    const int wid  = tid >> 5;
    if (lane == 0) wred[wid] = prod;
    __syncthreads();
    if (tid == 0) {
        float p = 1.0f;
        const int nw = (nt + 31) >> 5;
        for (int i = 0; i < nw; ++i) p *= wred[i];
        out[b] = p;
    }
}

extern "C" void kernel_launch(void* const* d_in, const int* in_sizes, int n_in,
                              void* d_out, int out_size, void* d_ws, size_t ws_size,
                              hipStream_t stream) {
    const float* z     = (const float*)d_in[0];   // (B, M, S) f32
    const float* mT    = (const float*)d_in[1];   // (B, M)    f32
    const int*   obs   = (const int*)d_in[2];     // (B, 2)    int
    const float* pa    = (const float*)d_in[3];   // scalar
    const float* pb    = (const float*)d_in[4];   // scalar
    const float* pbase = (const float*)d_in[5];   // (2,)
    float* out = (float*)d_out;

    const int B = out_size;                       // one weight per event
    const int M = in_sizes[1] / B;                // 128
    const int S = in_sizes[0] / in_sizes[1];      // 25 (1 accepted + 24 rejected)
    const int total = M * S;

    int threads = ((M + 31) / 32) * 32;           // one thread per row m
    if (threads < 64) threads = 64;
    if (threads > 256) threads = 256;

    size_t smem = (size_t)total * sizeof(float);  // ~12.8 KB of the 320 KB/WGP LDS

    lund_weight_kernel<<<B, threads, smem, stream>>>(z, mT, obs, pa, pb, pbase,
                                                     out, M, S);
}